// DualHOILoss_70291434766589
// MI455X (gfx1250) — compile-verified
//
#include <hip/hip_runtime.h>
#include <hip/hip_bf16.h>
#include <math.h>

// Problem constants from the reference
#define B_  16
#define P_  4096
#define A_  32
#define V_  778
#define WPB 8                        // waves per block (8 * 32 = 256 threads)
#define NTILES (B_ * (P_ / 16))      // 4096 point-tiles, one per wave
#define VTILES ((V_ + 15) / 16)      // 49 vert tiles
#define VPAD   (VTILES * 16)         // 784 padded vertex slots
#define VDW    (V_ * 3)              // 2334 real vert dwords
#define VDWP   (VPAD * 3)            // 2352 staged dwords (raw strip)

typedef __attribute__((ext_vector_type(2))) float v2f;
typedef __attribute__((ext_vector_type(8))) float v8f;

typedef __attribute__((address_space(1))) int  gint;   // global int
typedef __attribute__((address_space(3))) int  sint;   // LDS int

__global__ void hoi_init(float* accum) {
    accum[0] = 0.0f;
    accum[1] = 0.0f;
}

__global__ __launch_bounds__(256) void hoi_main(
    const float* __restrict__ verts,    // (B, V, 3)
    const float* __restrict__ anchors,  // (B, A, 3)
    const float* __restrict__ choir,    // (B, P, 6)
    const float* __restrict__ hc,       // (B, P)
    const float* __restrict__ bmean,    // (B, 1, 3)
    const float* __restrict__ bscalar,  // (1,)
    const float* __restrict__ basis,    // (P, 3)
    float* __restrict__ accum)          // [0]=choir sum, [1]=contact sum
{
    __shared__ float rawv[VDWP];        // raw packed x,y,z strip (async landing zone)
    __shared__ float v4[VPAD * 4];      // (x, y, z, 0) per vertex, 16B aligned
    __shared__ float vnrm[VPAD];        // |vert|^2, 1e30 poison in pad slots
    __shared__ float sC[WPB], sH[WPB];

    const int tid  = threadIdx.x;
    const int lane = tid & 31;
    const int wav  = tid >> 5;
    const int gw   = blockIdx.x * WPB + wav;   // global wave = tile id
    const int b    = gw >> 8;                  // 256 tiles per batch; uniform in block
    const int tile = gw & 255;
    const int pBase = tile * 16;
    const int m    = lane & 15;                // row/col within tile
    const int hi   = lane >> 4;                // 0: (x,y) half, 1: (z,0) half
    const int p    = pBase + m;

    // ---- stage verts[b] into LDS once per block ----
    {
        const float* gsrc = verts + (long)b * V_ * 3;
#if __has_builtin(__builtin_amdgcn_global_load_async_to_lds_b32)
        for (int i = tid; i < VDW; i += 256) {
            __builtin_amdgcn_global_load_async_to_lds_b32(
                (gint*)(gsrc + i), (sint*)&rawv[i], /*offset=*/0, /*cpol=*/0);
        }
#if __has_builtin(__builtin_amdgcn_s_wait_asynccnt)
        __builtin_amdgcn_s_wait_asynccnt(0);
#else
        asm volatile("s_wait_asynccnt 0x0" ::: "memory");
#endif
#else
        for (int i = tid; i < VDW; i += 256) rawv[i] = gsrc[i];
#endif
        for (int i = VDW + tid; i < VDWP; i += 256) rawv[i] = 0.0f;
    }
    __syncthreads();

    // Repack: (x,y,z,0) fragments + precomputed norms (poisoned tail)
    for (int v = tid; v < VPAD; v += 256) {
        const float vx = rawv[v * 3 + 0];
        const float vy = rawv[v * 3 + 1];
        const float vz = rawv[v * 3 + 2];
        v4[v * 4 + 0] = vx;
        v4[v * 4 + 1] = vy;
        v4[v * 4 + 2] = vz;
        v4[v * 4 + 3] = 0.0f;
        vnrm[v] = (v < V_) ? (vx * vx + vy * vy + vz * vz) : 1e30f;
    }
    __syncthreads();

    const float inv = 1.0f / bscalar[0];
    const float mx = bmean[b * 3 + 0];
    const float my = bmean[b * 3 + 1];
    const float mz = bmean[b * 3 + 2];

    // Decode target point for row m of this tile (both halves compute it)
    const long cbase = ((long)(b * P_) + p) * 6;
    const float dx = choir[cbase + 1];
    const float dy = choir[cbase + 2];
    const float dz = choir[cbase + 3];
    const float bx = basis[p * 3 + 0];
    const float by = basis[p * 3 + 1];
    const float bz = basis[p * 3 + 2];
    const float tx = (bx + dx) * inv + mx;
    const float ty = (by + dy) * inv + my;
    const float tz = (bz + dz) * inv + mz;
    const float nx = tx * tx + ty * ty + tz * tz;

    // A fragment: 16x4 f32, K=3 padded. lanes0-15: K0,K1 = x,y; lanes16-31: K2,K3 = z,0
    v2f a;
    a.x = hi ? tz : tx;
    a.y = hi ? 0.0f : ty;

    // Row norms per C register r (uniform across each 16-lane half) — added AFTER the
    // min reduction: min_j(nrow + nv - 2 dot) = nrow + min_j(nv - 2 dot)
    float nrow[8];
#pragma unroll
    for (int r = 0; r < 8; ++r) nrow[r] = __shfl(nx, r + 8 * hi, 32);

    float rmin[8];
#pragma unroll
    for (int r = 0; r < 8; ++r) rmin[r] = 1e30f;

    for (int j = 0; j < VTILES; ++j) {
        const int v = j * 16 + m;
        // B fragment directly from LDS: 8B at +hi*8 gives (x,y) or (z,0)
        const v2f bf = *(const v2f*)&v4[(unsigned)v * 4u + (unsigned)hi * 2u];
        const float nv = vnrm[v];

        v8f c = {};
        c = __builtin_amdgcn_wmma_f32_16x16x4_f32(
                /*neg_a=*/false, a, /*neg_b=*/false, bf,
                /*c_mod=*/(short)0, c, /*reuse_a=*/false, /*reuse_b=*/false);

#pragma unroll
        for (int r = 0; r < 8; ++r) {
            const float t = __builtin_fmaf(-2.0f, c[r], nv);  // nv - 2*dot
            rmin[r] = fminf(rmin[r], t);
        }
    }

    // Column-min butterfly inside each 16-lane half, then contact loss term
    float sumH = 0.0f;
#pragma unroll
    for (int r = 0; r < 8; ++r) {
        float mn = rmin[r];
        mn = fminf(mn, __shfl_xor(mn, 1, 32));
        mn = fminf(mn, __shfl_xor(mn, 2, 32));
        mn = fminf(mn, __shfl_xor(mn, 4, 32));
        mn = fminf(mn, __shfl_xor(mn, 8, 32));
        mn += nrow[r];                      // re-add row norm: full min d^2
        if (m == 0) {                       // lanes 0 and 16 own rows r and r+8
            const int row = r + 8 * hi;
            const float contact = __expf(-100.0f * fmaxf(mn, 1e-12f));
            const float h = hc[b * P_ + pBase + row];
            const float e = h - contact;
            sumH += e * e;
        }
    }

    // Choir loss: distance to the single gathered anchor (lanes 0-15 only)
    float sumC = 0.0f;
    if (!hi) {
        const int idx = (int)choir[cbase + 5];
        const float* ap = anchors + ((long)(b * A_ + idx)) * 3;
        const float ex = tx - ap[0];
        const float ey = ty - ap[1];
        const float ez = tz - ap[2];
        const float d2 = ex * ex + ey * ey + ez * ez;
        const float d  = sqrtf(fmaxf(d2, 1e-12f));
        const float e  = d - choir[cbase + 4];
        sumC = e * e;
    }

    // Full-wave reduction of both partial sums
#pragma unroll
    for (int mask = 16; mask >= 1; mask >>= 1) {
        sumH += __shfl_xor(sumH, mask, 32);
        sumC += __shfl_xor(sumC, mask, 32);
    }
    if (lane == 0) { sC[wav] = sumC; sH[wav] = sumH; }
    __syncthreads();
    if (tid == 0) {
        float tc = 0.0f, th = 0.0f;
        for (int w = 0; w < WPB; ++w) { tc += sC[w]; th += sH[w]; }
        atomicAdd(&accum[0], tc);
        atomicAdd(&accum[1], th);
    }
}

__global__ void hoi_finalize(const float* __restrict__ accum, float* __restrict__ out) {
    const float n = (float)(B_ * P_);
    out[0] = accum[0] / n;   // choir_loss
    out[1] = accum[1] / n;   // hand_contact_loss
}

extern "C" void kernel_launch(void* const* d_in, const int* in_sizes, int n_in,
                              void* d_out, int out_size, void* d_ws, size_t ws_size,
                              hipStream_t stream) {
    const float* verts   = (const float*)d_in[0];
    const float* anchors = (const float*)d_in[1];
    const float* choir   = (const float*)d_in[2];
    const float* hc      = (const float*)d_in[3];
    const float* bmean   = (const float*)d_in[4];
    const float* bscalar = (const float*)d_in[5];
    const float* basis   = (const float*)d_in[6];
    float* out   = (float*)d_out;
    float* accum = (float*)d_ws;

    hoi_init<<<1, 1, 0, stream>>>(accum);
    hoi_main<<<NTILES / WPB, 256, 0, stream>>>(verts, anchors, choir, hc,
                                               bmean, bscalar, basis, accum);
    hoi_finalize<<<1, 1, 0, stream>>>(accum, out);
}